// Exchange_2104533975589
// MI455X (gfx1250) — compile-verified
//
#include <hip/hip_runtime.h>

// ---------------------------------------------------------------------------
// Channel-exchange (8,512,4096 fp32 x2).  Memory-bound permuted copy:
// 268 MB moved -> ~11.5 us at 23.3 TB/s HBM.  No matrix math, so the CDNA5
// feature of interest is the async global<->LDS DMA path (ASYNCcnt).
// ---------------------------------------------------------------------------

#define MAXC 1024

// One block, C threads.  Computes, for every output channel of y1 and y2,
// the source (which input tensor, which channel).  Exactly matches
// jax.lax.top_k tie semantics: rank by (|bn| desc, index asc).
__global__ void build_maps_kernel(const float* __restrict__ bn1,
                                  const float* __restrict__ bn2,
                                  int* __restrict__ maps, int C, int topk) {
    __shared__ float a1[MAXC], a2[MAXC];
    __shared__ unsigned char m1[MAXC], m2[MAXC];   // 1 = non-top
    __shared__ short nt1[MAXC], nt2[MAXC];         // non-top channel lists
    const int i = threadIdx.x;
    a1[i] = fabsf(bn1[i]);
    a2[i] = fabsf(bn2[i]);
    __syncthreads();

    const float v1i = a1[i], v2i = a2[i];
    int r1 = 0, r2 = 0;
    for (int j = 0; j < C; ++j) {
        const float v1 = a1[j], v2 = a2[j];
        r1 += (v1 > v1i) || (v1 == v1i && j < i);
        r2 += (v2 > v2i) || (v2 == v2i && j < i);
    }
    const bool top1 = r1 < topk, top2 = r2 < topk;
    m1[i] = !top1;
    m2[i] = !top2;
    __syncthreads();

    // position of channel i within the ascending-ordered non-top list
    int p1 = 0, p2 = 0;
    for (int j = 0; j < i; ++j) { p1 += m1[j]; p2 += m2[j]; }
    if (!top1) nt1[p1] = (short)i;
    if (!top2) nt2[p2] = (short)i;
    __syncthreads();

    // encoding: bit16 = read from x2, low 16 bits = source channel
    // y1[:,c,:] = top1(c) ? x1[:,c,:] : x2[:, nontop2[pos1(c)], :]
    // y2[:,c,:] = top2(c) ? x2[:,c,:] : x1[:, nontop1[pos2(c)], :]
    maps[i]     = top1 ? i                : ((1 << 16) | (int)nt2[p1]);
    maps[C + i] = top2 ? ((1 << 16) | i)  : ((int)nt1[p2]);
}

// One block moves one 4096-float (16 KiB) row chunk through LDS using the
// CDNA5 async global->LDS->global path.  256 threads x 4 float4 each.
// Each thread stores exactly the LDS bytes it loaded, so only a per-wave
// s_wait_asynccnt is needed between the two phases (no barrier).
__global__ void __launch_bounds__(256)
exchange_kernel(const float* __restrict__ x1,
                const float* __restrict__ x2,
                const int* __restrict__ maps,
                float* __restrict__ out,
                int B, int C, int L, int chunks_per_row) {
    const unsigned blk = blockIdx.x;
    const unsigned row = blk / chunks_per_row;
    const unsigned chunk = blk - row * chunks_per_row;

    const unsigned bc = (unsigned)B * (unsigned)C;
    const unsigned t = row / bc;            // 0 = y1, 1 = y2
    const unsigned rem = row - t * bc;
    const unsigned b = rem / (unsigned)C;
    const unsigned c = rem - b * (unsigned)C;

    const int m = maps[t * C + c];
    const float* src = (m & (1 << 16)) ? x2 : x1;
    const unsigned sc = (unsigned)(m & 0xFFFF);

    const float4* __restrict__ srow =
        (const float4*)(src + ((size_t)b * C + sc) * (size_t)L) +
        (size_t)chunk * 1024u;
    float4* __restrict__ drow =
        (float4*)(out + (((size_t)t * B + b) * C + c) * (size_t)L) +
        (size_t)chunk * 1024u;

    __shared__ float4 tile[1024];           // 16 KiB per workgroup
    const unsigned tid = threadIdx.x;

    // Phase 1: async global -> LDS (tracked by ASYNCcnt)
#pragma unroll
    for (int k = 0; k < 4; ++k) {
        const unsigned e = (unsigned)k * 256u + tid;
        // generic pointer to shared: low 32 bits == workgroup LDS byte offset
        const unsigned lds_addr = (unsigned)(unsigned long long)(&tile[e]);
        const unsigned long long ga = (unsigned long long)(const void*)(srow + e);
        asm volatile("global_load_async_to_lds_b128 %0, %1, off"
                     :
                     : "v"(lds_addr), "v"(ga)
                     : "memory");
    }
    asm volatile("s_wait_asynccnt 0" ::: "memory");

    // Phase 2: async LDS -> global
#pragma unroll
    for (int k = 0; k < 4; ++k) {
        const unsigned e = (unsigned)k * 256u + tid;
        const unsigned lds_addr = (unsigned)(unsigned long long)(&tile[e]);
        const unsigned long long ga = (unsigned long long)(void*)(drow + e);
        asm volatile("global_store_async_from_lds_b128 %0, %1, off"
                     :
                     : "v"(ga), "v"(lds_addr)
                     : "memory");
    }
    // S_ENDPGM implies wait-idle, but make the dependency explicit.
    asm volatile("s_wait_asynccnt 0" ::: "memory");
}

extern "C" void kernel_launch(void* const* d_in, const int* in_sizes, int n_in,
                              void* d_out, int out_size, void* d_ws, size_t ws_size,
                              hipStream_t stream) {
    const float* x1  = (const float*)d_in[0];
    const float* x2  = (const float*)d_in[1];
    const float* bn1 = (const float*)d_in[2];
    const float* bn2 = (const float*)d_in[3];

    const int C = in_sizes[2];                     // 512
    const int B = 8;                               // fixed by reference setup
    const long long total = (long long)in_sizes[0];
    const int L = (int)(total / ((long long)B * C));   // 4096
    const int topk = C / 2;                        // EXCHANGE_RATIO = 0.5

    int* maps = (int*)d_ws;                        // 2*C ints of scratch

    build_maps_kernel<<<1, C, 0, stream>>>(bn1, bn2, maps, C, topk);

    const int chunks_per_row = (L / 4) / 1024;     // 1 for L=4096
    const unsigned rows = 2u * (unsigned)B * (unsigned)C;
    exchange_kernel<<<dim3(rows * (unsigned)chunks_per_row), 256, 0, stream>>>(
        x1, x2, maps, (float*)d_out, B, C, L, chunks_per_row);
}